// GPR_sparse_28192165331246
// MI455X (gfx1250) — compile-verified
//
#include <hip/hip_runtime.h>
#include <hip/hip_bf16.h>

// GPR-GNN forward: hidden = x*temp[0] + sum_i relu(scatter(gather(h@W_i^T+b_i)*ew)) * temp[i+1]
// D = 128 fixed. fp32 throughout; dense GEMM uses V_WMMA_F32_16X16X4_F32.

#define DIMS 128

typedef __attribute__((ext_vector_type(2))) float v2f;
typedef __attribute__((ext_vector_type(8))) float v8f;

// ---------------------------------------------------------------------------
// init: h = x (pad rows zeroed), out = x*temp[0], t2 = 0  (float4 per thread)
// ---------------------------------------------------------------------------
__global__ __launch_bounds__(256) void k_init(const float* __restrict__ x,
                                              const float* __restrict__ temp,
                                              float* __restrict__ h,
                                              float* __restrict__ t2,
                                              float* __restrict__ out,
                                              int nElem, int nElemPad) {
    int vid = (blockIdx.x * 256 + threadIdx.x) * 4;
    if (vid >= nElemPad) return;
    *(float4*)(t2 + vid) = make_float4(0.f, 0.f, 0.f, 0.f);
    if (vid < nElem) {
        float4 xv = *(const float4*)(x + vid);
        float  t0 = temp[0];
        *(float4*)(h + vid)   = xv;
        *(float4*)(out + vid) = make_float4(xv.x * t0, xv.y * t0, xv.z * t0, xv.w * t0);
    } else {
        *(float4*)(h + vid) = make_float4(0.f, 0.f, 0.f, 0.f);
    }
}

// ---------------------------------------------------------------------------
// GEMM: t1[n, :] = h[n, :] @ W^T + b   (W is [128 out][128 in], row-major)
// Block = 256 threads = 8 waves; block covers 128 node rows; wave w covers
// rows [blk*128 + 16w, +16) x all 8 column tiles.
// W staged into LDS PAIR-INTERLEAVED: lds[(k>>1)*256 + n*2 + (k&1)] = W[n][k],
// so each B fragment is a single aligned ds_load_b64 (no repack movs).
// A loads are software-pipelined one K-step ahead.
// ---------------------------------------------------------------------------
__global__ __launch_bounds__(256) void k_gemm(const float* __restrict__ h,
                                              const float* __restrict__ W,
                                              const float* __restrict__ bias,
                                              float* __restrict__ t1,
                                              int nNodes) {
    __shared__ float Wt[DIMS * DIMS];
    const int t = threadIdx.x;

    // stage weights pair-interleaved: coalesced global reads
    #pragma unroll
    for (int it = 0; it < (DIMS * DIMS) / 256; ++it) {
        int idx = it * 256 + t;
        int n = idx >> 7;
        int k = idx & 127;
        Wt[(k >> 1) * (2 * DIMS) + n * 2 + (k & 1)] = W[idx];
    }
    __syncthreads();

    const int wave = t >> 5;
    const int lane = t & 31;
    const int l15  = lane & 15;
    const int hi   = (lane >> 4) & 1;           // selects K pair / output row half
    const long rowBase = (long)blockIdx.x * 128 + wave * 16;

    // C init with broadcast bias: C[m][n] = b[n]; slot (v,lane) has N = lane&15
    v8f acc[8];
    #pragma unroll
    for (int j = 0; j < 8; ++j) {
        float bv = bias[j * 16 + l15];
        #pragma unroll
        for (int v = 0; v < 8; ++v) acc[j][v] = bv;
    }

    // A-fragment row for this lane (padded buffer -> always in-bounds)
    const float* hrow = h + (rowBase + l15) * (long)DIMS;

    v2f a = *(const v2f*)(hrow + hi * 2);       // prefetch K-step 0
    for (int k0 = 0; k0 < DIMS; k0 += 4) {
        const v2f acur = a;
        if (k0 + 4 < DIMS)                      // prefetch next K-step
            a = *(const v2f*)(hrow + (k0 + 4) + hi * 2);
        const int kk = k0 + hi * 2;
        const v2f* bpairs = (const v2f*)(Wt + (kk >> 1) * (2 * DIMS));
        #pragma unroll
        for (int j = 0; j < 8; ++j) {
            v2f bf = bpairs[j * 16 + l15];      // single ds_load_b64
            acc[j] = __builtin_amdgcn_wmma_f32_16x16x4_f32(
                false, acur, false, bf, (short)0, acc[j], false, false);
        }
    }

    // store D: slot v -> row rowBase + v + 8*hi, col j*16 + (lane&15)
    #pragma unroll
    for (int j = 0; j < 8; ++j) {
        #pragma unroll
        for (int v = 0; v < 8; ++v) {
            long row = rowBase + v + hi * 8;
            if (row < nNodes) t1[row * DIMS + j * 16 + l15] = acc[j][v];
        }
    }
}

// ---------------------------------------------------------------------------
// edge scatter: t2[dst] += t1[src] * ew   (one wave per edge; lane owns 4 floats)
// ~410M L2-resident global_atomic_add_f32 total -> the roofline-dominant pass.
// ---------------------------------------------------------------------------
__global__ __launch_bounds__(256) void k_scatter(const float* __restrict__ t1,
                                                 const float* __restrict__ ew,
                                                 const int* __restrict__ esrc,
                                                 const int* __restrict__ edst,
                                                 float* __restrict__ t2,
                                                 int E) {
    long gid  = (long)blockIdx.x * 256 + threadIdx.x;
    int  edge = (int)(gid >> 5);
    int  lane = (int)(gid & 31);
    if (edge >= E) return;
    int   s = esrc[edge];
    int   d = edst[edge];
    float w = ew[edge];
    float4 v = *(const float4*)(t1 + (long)s * DIMS + lane * 4);
    float* base = t2 + (long)d * DIMS + lane * 4;
    unsafeAtomicAdd(base + 0, v.x * w);   // global_atomic_add_f32
    unsafeAtomicAdd(base + 1, v.y * w);
    unsafeAtomicAdd(base + 2, v.z * w);
    unsafeAtomicAdd(base + 3, v.w * w);
}

// ---------------------------------------------------------------------------
// finish: h = relu(t2); out += h*temp[layer+1]; t2 = 0 for next layer's scatter
// (fuses the zero pass -> saves a full 25.6 MB stream per layer)
// ---------------------------------------------------------------------------
__global__ __launch_bounds__(256) void k_finish(float* __restrict__ t2,
                                                const float* __restrict__ temp,
                                                float* __restrict__ h,
                                                float* __restrict__ out,
                                                int layer, int nElem, int nElemPad) {
    int vid = (blockIdx.x * 256 + threadIdx.x) * 4;
    if (vid >= nElemPad) return;
    float4 v = *(const float4*)(t2 + vid);
    *(float4*)(t2 + vid) = make_float4(0.f, 0.f, 0.f, 0.f);
    v.x = v.x > 0.f ? v.x : 0.f;
    v.y = v.y > 0.f ? v.y : 0.f;
    v.z = v.z > 0.f ? v.z : 0.f;
    v.w = v.w > 0.f ? v.w : 0.f;
    *(float4*)(h + vid) = v;
    if (vid < nElem) {
        float  tc = temp[layer + 1];
        float4 o  = *(const float4*)(out + vid);
        o.x += v.x * tc; o.y += v.y * tc; o.z += v.z * tc; o.w += v.w * tc;
        *(float4*)(out + vid) = o;
    }
}

// ---------------------------------------------------------------------------
extern "C" void kernel_launch(void* const* d_in, const int* in_sizes, int n_in,
                              void* d_out, int out_size, void* d_ws, size_t ws_size,
                              hipStream_t stream) {
    (void)n_in; (void)out_size; (void)ws_size;

    const float* x      = (const float*)d_in[0];
    const float* edge_w = (const float*)d_in[1];
    const float* W      = (const float*)d_in[2];
    const float* b      = (const float*)d_in[3];
    const float* temp   = (const float*)d_in[4];
    const int*   esrc   = (const int*)d_in[5];
    const int*   edst   = (const int*)d_in[6];
    float*       out    = (float*)d_out;

    const int N  = in_sizes[0] / DIMS;           // 50000
    const int E  = in_sizes[1];                  // 800000
    const int L  = in_sizes[2] / (DIMS * DIMS);  // 4
    const int NP = ((N + 127) / 128) * 128;      // padded rows for WMMA tiles
    const int nElem    = N * DIMS;
    const int nElemPad = NP * DIMS;

    // workspace: h | t1 | t2  (each NP*D floats; ~77 MB total)
    float* h  = (float*)d_ws;
    float* t1 = h  + (size_t)nElemPad;
    float* t2 = t1 + (size_t)nElemPad;

    const int ewBlocks   = (nElemPad / 4 + 255) / 256;
    const int gemmBlocks = NP / 128;
    const int scBlocks   = (E * 32 + 255) / 256;

    k_init<<<ewBlocks, 256, 0, stream>>>(x, temp, h, t2, out, nElem, nElemPad);

    for (int i = 0; i < L; ++i) {
        k_gemm<<<gemmBlocks, 256, 0, stream>>>(h, W + (size_t)i * DIMS * DIMS,
                                               b + (size_t)i * DIMS, t1, N);
        k_scatter<<<scBlocks, 256, 0, stream>>>(t1, edge_w, esrc, edst, t2, E);
        k_finish<<<ewBlocks, 256, 0, stream>>>(t2, temp, h, out, i, nElem, nElemPad);
    }
}